// Att_fusion_final_47502338294114
// MI455X (gfx1250) — compile-verified
//
#include <hip/hip_runtime.h>
#include <math.h>

typedef float v2f __attribute__((ext_vector_type(2)));
typedef float v8f __attribute__((ext_vector_type(8)));
typedef unsigned int u32x4 __attribute__((ext_vector_type(4)));
typedef int i32x4 __attribute__((ext_vector_type(4)));
typedef int i32x8 __attribute__((ext_vector_type(8)));

#define N_    16
#define C_    64
#define WH_   16384

// ---- workspace layout (float offsets) ----
#define WS_G    0                 // [16][2][64][64] gram accumulators
#define WS_S    131072            // [16][2][64]     channel sums
#define WS_ATT  133120            // [16][64][128]   att^T  (out-ch major, k contiguous)
#define WS_ZERO 133120            // floats to zero each launch (G + S)

// ------------------------------------------------------------------
// Kernel 0: zero the atomic-accumulated region (must happen per launch)
// ------------------------------------------------------------------
__global__ void zero_ws(float* __restrict__ ws, int n) {
    int i = blockIdx.x * blockDim.x + threadIdx.x;
    if (i < n) ws[i] = 0.0f;
}

// ------------------------------------------------------------------
// Kernel 1: per (n, modality, k-chunk) partial channel Gram G = X X^T
// and channel sums s = X 1, fp32 WMMA 16x16x4.
// Block: 512 threads = 16 waves; wave w owns G tile (mi=w>>2, ni=w&3).
// ------------------------------------------------------------------
#define K1_CHUNK  1024      // pixels per block
#define K1_STAGE  128       // pixels staged in LDS per iteration
#define K1_LP     132       // padded LDS pixel stride (floats)

__global__ __launch_bounds__(512) void gram_kernel(
        const float* __restrict__ xR, const float* __restrict__ xT,
        float* __restrict__ ws) {
    __shared__ float xs[C_ * K1_LP];
    __shared__ float psum[512];

    int blk   = blockIdx.x;          // 16 n * 2 mod * 16 chunks
    int chunk = blk & 15;
    int mod   = (blk >> 4) & 1;
    int n     = blk >> 5;
    const float* X = (mod ? xT : xR) + (size_t)n * C_ * WH_;
    int pixbase = chunk * K1_CHUNK;

    int t = threadIdx.x;
    int wave = t >> 5, lane = t & 31;
    int mi = wave >> 2, ni = wave & 3;
    int arow = mi * 16 + (lane & 15);
    int brow = ni * 16 + (lane & 15);
    int koff = (lane >> 4) * 2;      // low half-wave K={k,k+1}, high K={k+2,k+3}

    // cooperative-load role: channel t>>3, 16-pixel run (t&7)*16
    int lch = t >> 3;
    int lpx = (t & 7) * 16;

    float csum = 0.0f;
    v8f acc = {};

    for (int st = 0; st < K1_CHUNK / K1_STAGE; ++st) {
        __syncthreads();
        const float* src = X + (size_t)lch * WH_ + pixbase + st * K1_STAGE + lpx;
        float* dst = &xs[lch * K1_LP + lpx];
        #pragma unroll
        for (int i = 0; i < 16; i += 4) {
            float4 v = *(const float4*)(src + i);
            *(float4*)(dst + i) = v;
            csum += v.x + v.y + v.z + v.w;
        }
        __syncthreads();
        #pragma unroll
        for (int k = 0; k < K1_STAGE; k += 4) {
            v2f A = *(const v2f*)&xs[arow * K1_LP + k + koff];
            v2f B = *(const v2f*)&xs[brow * K1_LP + k + koff];
            acc = __builtin_amdgcn_wmma_f32_16x16x4_f32(
                    false, A, false, B, (short)0, acc, false, false);
        }
    }

    // accumulate this block's partial G tile
    float* G = ws + WS_G + (size_t)(n * 2 + mod) * 4096;
    int col   = ni * 16 + (lane & 15);
    int rbase = mi * 16 + (lane >> 4) * 8;
    #pragma unroll
    for (int v = 0; v < 8; ++v)
        unsafeAtomicAdd(&G[(rbase + v) * 64 + col], acc[v]);

    // channel sums: reduce 8 threads per channel
    psum[t] = csum;
    __syncthreads();
    if (t < 64) {
        float s = 0.0f;
        #pragma unroll
        for (int i = 0; i < 8; ++i) s += psum[t * 8 + i];
        unsafeAtomicAdd(&ws[WS_S + (n * 2 + mod) * 64 + t], s);
    }
}

// ------------------------------------------------------------------
// Kernel 2: per-n tiny math: rs = W G W^T + u b^T + b u^T + WH b b^T,
// logits [128][64], column softmax, store att^T [64][128].
// Block: 256 threads, one block per n. Dynamic LDS: W|T1|L|u|b.
// ------------------------------------------------------------------
__global__ __launch_bounds__(256) void att_kernel(
        const float* __restrict__ WR, const float* __restrict__ bR,
        const float* __restrict__ WT, const float* __restrict__ bT,
        float* __restrict__ ws) {
    extern __shared__ float sm[];
    float* Wsh = sm;            // 4096
    float* T1  = sm + 4096;     // 4096
    float* L   = sm + 8192;     // 8192  logits [128][64]
    float* u   = sm + 16384;    // 64
    float* bb  = sm + 16448;    // 64

    int n = blockIdx.x, t = threadIdx.x;

    for (int mod = 0; mod < 2; ++mod) {
        const float* Wm = mod ? WT : WR;
        const float* bm = mod ? bT : bR;
        const float* G  = ws + WS_G + (size_t)(n * 2 + mod) * 4096;

        __syncthreads();   // protect LDS reuse across mod iterations
        for (int i = t; i < 4096; i += 256) Wsh[i] = Wm[i];
        if (t < 64) bb[t] = bm[t];
        __syncthreads();

        if (t < 64) {
            const float* sv = ws + WS_S + (n * 2 + mod) * 64;
            float s = 0.0f;
            for (int e = 0; e < 64; ++e) s += Wsh[t * 64 + e] * sv[e];
            u[t] = s;
        }
        // T1 = W @ G   (G streamed from L2)
        for (int i = t; i < 4096; i += 256) {
            int r = i >> 6, c = i & 63;
            float s = 0.0f;
            for (int e = 0; e < 64; ++e) s += Wsh[r * 64 + e] * G[e * 64 + c];
            T1[i] = s;
        }
        __syncthreads();
        // M1 = T1 @ W^T  plus rank-1 bias terms -> logits
        for (int i = t; i < 4096; i += 256) {
            int r = i >> 6, c = i & 63;
            float s = 0.0f;
            for (int e = 0; e < 64; ++e) s += T1[r * 64 + e] * Wsh[c * 64 + e];
            s += u[r] * bb[c] + bb[r] * u[c] + (float)WH_ * bb[r] * bb[c];
            L[(mod * 64 + r) * 64 + c] = s;
        }
    }
    __syncthreads();

    if (t < 64) {  // softmax over the 128 rows of column t
        float mx = -3.0e38f;
        for (int k = 0; k < 128; ++k) mx = fmaxf(mx, L[k * 64 + t]);
        float sum = 0.0f;
        for (int k = 0; k < 128; ++k) {
            float e = expf(L[k * 64 + t] - mx);
            L[k * 64 + t] = e;
            sum += e;
        }
        float inv = 1.0f / sum;
        float* attT = ws + WS_ATT + (size_t)n * 8192 + t * 128;
        for (int k = 0; k < 128; ++k) attT[k] = L[k * 64 + t] * inv;
    }
}

// ------------------------------------------------------------------
// Kernel 3: out[n] (64 x WH) = attT (64x128) @ concat(xR[n],xT[n]) (128 x WH)
// Block: 512 threads, 256-pixel chunk. X tile staged in dynamic LDS via the
// Tensor Data Mover (tensor_load_to_lds): one 64x256 f32 tile per modality,
// LDS-pad 8 DWORDs every 256 DWORDs -> row stride 264 floats (bank-safe).
// Wave w: M-tile (w&3), N-tiles (w>>2)*4 .. +3 (A reused 4x per step).
// ------------------------------------------------------------------
#define K3_PIX 256
#define K3_LP  264        // 256 data + 8 pad floats: 2*264 % 64 == 16 -> no conflicts

__device__ __forceinline__ void tdm_load_tile_f32(
        const float* gsrc, unsigned lds_byte_off) {
    // ---- D# group 0 ----
    unsigned long long ga = (unsigned long long)gsrc;   // byte address, 57-bit
    u32x4 g0;
    g0.x = 1u;                                  // count=1, user mode
    g0.y = lds_byte_off;                        // lds_addr (bytes)
    g0.z = (unsigned)(ga & 0xffffffffu);        // global_addr[31:0]
    g0.w = (unsigned)(ga >> 32) | (2u << 30);   // global_addr[56:32] | type=2
    // ---- D# group 1 ----
    // data_size=2 (4B), pad_enable=1, pad_interval=7 (256 DW), pad_amount=7 (8 DW)
    i32x8 g1;
    g1[0] = (2 << 16) | (1 << 20) | (7 << 22) | (7 << 25);
    g1[1] = (int)((unsigned)(WH_ & 0xffff) << 16);        // tensor_dim0[15:0] (=16384) at [63:48]
    g1[2] = (int)(((unsigned)WH_ >> 16) | (64u << 16));   // tensor_dim0[31:16] | tensor_dim1[15:0]=64
    g1[3] = (int)((unsigned)K3_PIX << 16);                // tensor_dim1[31:16]=0 | tile_dim0=256
    g1[4] = 64;                                           // tile_dim1=64 rows, tile_dim2=0
    g1[5] = WH_;                                          // tensor_dim0_stride[31:0] = 16384
    g1[6] = 0;                                            // stride0[47:32]=0 | stride1[15:0]=0
    g1[7] = 0;
    i32x4 gz4 = {0, 0, 0, 0};                             // groups 2/3: unused (2-D tile)
    i32x8 gz8 = {0, 0, 0, 0, 0, 0, 0, 0};                 // extra operand (6-arg form)
    __builtin_amdgcn_tensor_load_to_lds(g0, g1, gz4, gz4, gz8, 0);
}

__global__ __launch_bounds__(512) void out_kernel(
        const float* __restrict__ xR, const float* __restrict__ xT,
        const float* __restrict__ ws, float* __restrict__ out) {
    extern __shared__ float xs[];   // [128][K3_LP]

    int blk = blockIdx.x;
    int n     = blk >> 6;           // 64 chunks per image
    int chunk = blk & 63;
    int pixbase = chunk * K3_PIX;
    int t = threadIdx.x;

    // ---- TDM: DMA both modality tiles into LDS (wave 0 issues) ----
    if (t == 0) {
        unsigned base = __builtin_amdgcn_groupstaticsize();   // dynamic LDS offset
        const float* srcR = xR + (size_t)n * 64 * WH_ + pixbase;
        const float* srcT = xT + (size_t)n * 64 * WH_ + pixbase;
        tdm_load_tile_f32(srcR, base);
        tdm_load_tile_f32(srcT, base + 64u * K3_LP * 4u);
        __builtin_amdgcn_s_wait_tensorcnt(0);
    }
    __syncthreads();

    int wave = t >> 5, lane = t & 31;
    int mi = wave & 3;
    int nb = (wave >> 2) * 4;
    const float* attT = ws + WS_ATT + (size_t)n * 8192;
    int arow = mi * 16 + (lane & 15);
    int koff = (lane >> 4) * 2;
    int cb   = nb * 16 + (lane & 15);

    v8f acc0 = {}, acc1 = {}, acc2 = {}, acc3 = {};
    #pragma unroll 8
    for (int k = 0; k < 128; k += 4) {
        v2f A = *(const v2f*)&attT[arow * 128 + k + koff];
        int r0 = (k + koff) * K3_LP;
        int r1 = r0 + K3_LP;
        v2f B0 = { xs[r0 + cb     ], xs[r1 + cb     ] };
        v2f B1 = { xs[r0 + cb + 16], xs[r1 + cb + 16] };
        v2f B2 = { xs[r0 + cb + 32], xs[r1 + cb + 32] };
        v2f B3 = { xs[r0 + cb + 48], xs[r1 + cb + 48] };
        acc0 = __builtin_amdgcn_wmma_f32_16x16x4_f32(false, A, false, B0, (short)0, acc0, false, false);
        acc1 = __builtin_amdgcn_wmma_f32_16x16x4_f32(false, A, false, B1, (short)0, acc1, false, false);
        acc2 = __builtin_amdgcn_wmma_f32_16x16x4_f32(false, A, false, B2, (short)0, acc2, false, false);
        acc3 = __builtin_amdgcn_wmma_f32_16x16x4_f32(false, A, false, B3, (short)0, acc3, false, false);
    }

    int col = lane & 15;
    int rb  = mi * 16 + (lane >> 4) * 8;
    float* outp = out + (size_t)n * 64 * WH_ + pixbase;
    #pragma unroll
    for (int v = 0; v < 8; ++v) {
        size_t ro = (size_t)(rb + v) * WH_;
        outp[ro + (nb + 0) * 16 + col] = acc0[v];
        outp[ro + (nb + 1) * 16 + col] = acc1[v];
        outp[ro + (nb + 2) * 16 + col] = acc2[v];
        outp[ro + (nb + 3) * 16 + col] = acc3[v];
    }
}

// ------------------------------------------------------------------
extern "C" void kernel_launch(void* const* d_in, const int* in_sizes, int n_in,
                              void* d_out, int out_size, void* d_ws, size_t ws_size,
                              hipStream_t stream) {
    const float* xR = (const float*)d_in[0];
    const float* xT = (const float*)d_in[1];
    const float* WR = (const float*)d_in[2];
    const float* bR = (const float*)d_in[3];
    const float* WT = (const float*)d_in[4];
    const float* bT = (const float*)d_in[5];
    float* out = (float*)d_out;
    float* ws  = (float*)d_ws;

    zero_ws<<<(WS_ZERO + 511) / 512, 512, 0, stream>>>(ws, WS_ZERO);
    gram_kernel<<<N_ * 2 * 16, 512, 0, stream>>>(xR, xT, ws);
    att_kernel<<<N_, 256, 16512 * sizeof(float), stream>>>(WR, bR, WT, bT, ws);
    out_kernel<<<N_ * 64, 512, 128 * K3_LP * sizeof(float), stream>>>(xR, xT, ws, out);
}